// MANN_Feature_36679020708360
// MI455X (gfx1250) — compile-verified
//
#include <hip/hip_runtime.h>

#define SLOTS 20
#define KDIM 64
#define LSTRIDE 68              // padded floats per slot row in LDS (bank-conflict-free)
#define WAVES_PER_BLOCK 8
#define THREADS (WAVES_PER_BLOCK * 32)

typedef __attribute__((ext_vector_type(2))) float v2f;
typedef __attribute__((ext_vector_type(4))) float v4f;
typedef __attribute__((ext_vector_type(8))) float v8f;

// ---- CDNA5 async global->LDS staging (ASYNCcnt-tracked), with safe fallback ----
#if defined(__has_builtin)
#if __has_builtin(__builtin_amdgcn_global_load_async_to_lds_b128) && \
    __has_builtin(__builtin_amdgcn_global_load_async_to_lds_b64)
#define USE_ASYNC_LDS 1
#endif
#endif
#ifndef USE_ASYNC_LDS
#define USE_ASYNC_LDS 0
#endif

// Pointee types per hipcc diagnostic: int __vector(4) / int __vector(2)
typedef int vi4 __attribute__((vector_size(16)));
typedef int vi2 __attribute__((vector_size(8)));
#define GAS4(p) ((__attribute__((address_space(1))) vi4*)(p))
#define LAS4(p) ((__attribute__((address_space(3))) vi4*)(p))
#define GAS2(p) ((__attribute__((address_space(1))) vi2*)(p))
#define LAS2(p) ((__attribute__((address_space(3))) vi2*)(p))

__global__ __launch_bounds__(THREADS) void mann_feature_kernel(
    const int* __restrict__ user,
    const float* __restrict__ query,
    const float* __restrict__ Value,
    float* __restrict__ out,
    int batch)
{
    __shared__ float lds_mk[WAVES_PER_BLOCK][SLOTS * LSTRIDE];
    __shared__ float lds_q [WAVES_PER_BLOCK][KDIM];
    __shared__ float lds_sc[WAVES_PER_BLOCK][32];

    const int lane = threadIdx.x & 31;
    const int wv   = threadIdx.x >> 5;
    const int b    = blockIdx.x * WAVES_PER_BLOCK + wv;
    const bool valid = (b < batch);

    // ---- Stage MK[user[b]] (20x64 f32, 5120B contiguous) + q[b] into padded LDS ----
    if (valid) {
        const int u = user[b];
        const float* __restrict__ src = Value + (size_t)u * (SLOTS * KDIM);
#if USE_ASYNC_LDS
        #pragma unroll
        for (int i = 0; i < 10; ++i) {
            int p4   = lane + 32 * i;      // float4 index 0..319, coalesced
            int row  = p4 >> 4;
            int col4 = p4 & 15;
            __builtin_amdgcn_global_load_async_to_lds_b128(
                GAS4(src + p4 * 4),
                LAS4(&lds_mk[wv][row * LSTRIDE + col4 * 4]), 0, 0);
        }
        __builtin_amdgcn_global_load_async_to_lds_b64(
            GAS2(query + (size_t)b * KDIM + 2 * lane),
            LAS2(&lds_q[wv][2 * lane]), 0, 0);
#else
        const v4f* __restrict__ src4 = (const v4f*)src;
        #pragma unroll
        for (int i = 0; i < 10; ++i) {
            int p4   = lane + 32 * i;
            int row  = p4 >> 4;
            int col4 = p4 & 15;
            v4f v = src4[p4];
            *(v4f*)&lds_mk[wv][row * LSTRIDE + col4 * 4] = v;
        }
        if (lane < 16) {
            *(v4f*)&lds_q[wv][lane * 4] =
                ((const v4f*)(query + (size_t)b * KDIM))[lane];
        }
#endif
    }
#if USE_ASYNC_LDS
#if __has_builtin(__builtin_amdgcn_s_wait_asynccnt)
    __builtin_amdgcn_s_wait_asynccnt(0);
#else
    asm volatile("s_wait_asynccnt 0" ::: "memory");
#endif
#endif
    __syncthreads();

    // ---- Scores: MK (20x64) @ q via V_WMMA_F32_16X16X4_F32 ----
    // A fragment: lane m=lane&15 holds MK[m, kbase+2h(+1)], h=lane>>4
    // B fragment: q replicated into all 16 N columns -> every lane gets scores in D
    // Tile1 rows are CLAMPED to slots 16..19 (finite dup data); D rows 20..31 unused.
    const int h  = lane >> 4;
    const int m  = lane & 15;
    const int r1 = 16 + (m & 3);

    v8f acc0 = {}; v8f acc1 = {};
    #pragma unroll
    for (int c = 0; c < 16; ++c) {
        const int k0 = 4 * c + 2 * h;
        v2f bq; bq.x = lds_q[wv][k0]; bq.y = lds_q[wv][k0 + 1];
        v2f a0 = *(const v2f*)&lds_mk[wv][m  * LSTRIDE + k0];
        v2f a1 = *(const v2f*)&lds_mk[wv][r1 * LSTRIDE + k0];
        acc0 = __builtin_amdgcn_wmma_f32_16x16x4_f32(false, a0, false, bq,
                                                     (short)0, acc0, false, false);
        acc1 = __builtin_amdgcn_wmma_f32_16x16x4_f32(false, a1, false, bq,
                                                     (short)0, acc1, false, false);
    }
    // D layout: vgpr r @ half h == scores[r + 8h] (identical across N columns)
    if (m == 0) {
        #pragma unroll
        for (int r = 0; r < 8; ++r) lds_sc[wv][8 * h + r] = acc0[r];
        if (h == 0) {
            #pragma unroll
            for (int r = 0; r < 4; ++r) lds_sc[wv][16 + r] = acc1[r];
        }
    }
    __syncthreads();

    if (!valid) return;

    // ---- Per-lane softmax(20) fused with weighted sum for k = 2*lane, 2*lane+1 ----
    float sc[SLOTS];
    float mx = -__builtin_inff();
    #pragma unroll
    for (int s = 0; s < SLOTS; ++s) {
        sc[s] = lds_sc[wv][s];
        mx = fmaxf(mx, sc[s]);
    }
    const int kk = 2 * lane;
    float p0 = 0.0f, p1 = 0.0f, sum = 0.0f;
    #pragma unroll
    for (int s = 0; s < SLOTS; ++s) {
        float e = __expf(sc[s] - mx);
        sum += e;
        v2f mk = *(const v2f*)&lds_mk[wv][s * LSTRIDE + kk];
        p0 = fmaf(e, mk.x, p0);
        p1 = fmaf(e, mk.y, p1);
    }
    const float inv = 1.0f / sum;
    v2f o; o.x = p0 * inv; o.y = p1 * inv;
    *(v2f*)&out[(size_t)b * KDIM + kk] = o;   // coalesced float2 store
}

extern "C" void kernel_launch(void* const* d_in, const int* in_sizes, int n_in,
                              void* d_out, int out_size, void* d_ws, size_t ws_size,
                              hipStream_t stream) {
    const int*   user  = (const int*)d_in[0];
    const float* query = (const float*)d_in[1];
    const float* Value = (const float*)d_in[2];
    float*       out   = (float*)d_out;
    const int batch = in_sizes[0];

    const int blocks = (batch + WAVES_PER_BLOCK - 1) / WAVES_PER_BLOCK;
    mann_feature_kernel<<<blocks, THREADS, 0, stream>>>(user, query, Value, out, batch);
}